// SharedDeformableLayer_57629871178414
// MI455X (gfx1250) — compile-verified
//
#include <hip/hip_runtime.h>
#include <hip/hip_bf16.h>
#include <math.h>

#define B_   8
#define C_   32
#define H_   256
#define W_   256
#define N_   (H_*W_)
#define CR_  2
#define EPS_ 1e-5f
#define CHUNK_ 2048            // LDS bitonic chunk (16 KB keys+vals)

typedef __attribute__((ext_vector_type(2))) float v2f;
typedef __attribute__((ext_vector_type(8))) float v8f;

__device__ __forceinline__ float gelu_erf(float x) {
    return 0.5f * x * (1.0f + erff(x * 0.70710678118654752f));
}

// ---------------- K1: per-(b,c) channel mean over H*W ----------------
__global__ void k_mean(const float* __restrict__ x, float* __restrict__ xc) {
    __shared__ float red[256];
    int bc = blockIdx.x;                       // b*C + c  (256 blocks)
    const float* p = x + (size_t)bc * N_;
    float s = 0.f;
    for (int i = threadIdx.x; i < N_; i += 256) s += p[i];
    red[threadIdx.x] = s;
    __syncthreads();
    for (int off = 128; off > 0; off >>= 1) {
        if (threadIdx.x < off) red[threadIdx.x] += red[threadIdx.x + off];
        __syncthreads();
    }
    if (threadIdx.x == 0) xc[bc] = red[0] * (1.0f / (float)N_);
}

// ---------------- K2: SE gate (C -> Cr -> C), one block of 256 -------
__global__ void k_gate(const float* __restrict__ xc,
                       const float* __restrict__ w1, const float* __restrict__ b1,
                       const float* __restrict__ w2, const float* __restrict__ b2,
                       float* __restrict__ sgate) {
    int t = threadIdx.x;                       // 256 = B*C
    int b = t >> 5, c = t & 31;
    float h[CR_];
    #pragma unroll
    for (int r = 0; r < CR_; ++r) {
        float a = b1[r];
        #pragma unroll
        for (int cc = 0; cc < C_; ++cc) a += xc[b*C_ + cc] * w1[r*C_ + cc];
        h[r] = gelu_erf(a);
    }
    float v = b2[c];
    #pragma unroll
    for (int r = 0; r < CR_; ++r) v += h[r] * w2[c*CR_ + r];
    sgate[t] = 1.0f / (1.0f + expf(-v));
}

// ---------------- K_rpe: resize 7x7 table -> C x 256 x 256 (half-pixel) ----
__global__ void k_rpe(const float* __restrict__ table, float* __restrict__ out) {
    int tid = blockIdx.x * 256 + threadIdx.x;  // C*N threads
    if (tid >= C_ * N_) return;
    int c = tid >> 16;
    int p = tid & (N_ - 1);
    int h = p >> 8, w = p & 255;
    float ys = fmaxf((h + 0.5f) * (7.0f / 256.0f) - 0.5f, 0.0f);
    float xs = fmaxf((w + 0.5f) * (7.0f / 256.0f) - 0.5f, 0.0f);
    int y0 = (int)floorf(ys), x0 = (int)floorf(xs);
    int y1 = min(y0 + 1, 6), x1 = min(x0 + 1, 6);
    float wy = ys - (float)y0, wx = xs - (float)x0;
    const float* tb = table + c * 49;
    float v00 = tb[y0*7+x0], v01 = tb[y0*7+x1], v10 = tb[y1*7+x0], v11 = tb[y1*7+x1];
    out[tid] = v00*(1.f-wy)*(1.f-wx) + v01*(1.f-wy)*wx + v10*wy*(1.f-wx) + v11*wy*wx;
}

// ---------------- K3: fused conv_offset with FP32 WMMA 1x1 conv ----------
// Block: 256 threads (8 waves), tile = 128 pixels of one row.
// Activation LDS tile 128 x 33 (stride 33 -> bank-conflict free).
// B matrix (conv2_w, zero-padded to 16 cols) staged in LDS once per block
// so the WMMA loop is pure ds_load + v_wmma, no exec-masked global loads.
__global__ void k_offset(const float* __restrict__ x,
                         const float* __restrict__ cw1, const float* __restrict__ cb1,
                         const float* __restrict__ sgate,
                         const float* __restrict__ lng, const float* __restrict__ lnb,
                         const float* __restrict__ w2,
                         float* __restrict__ posy, float* __restrict__ posx,
                         float* __restrict__ path) {
    __shared__ float lds[128 * 33];
    __shared__ float bpad[16 * 33];            // B[n][k], n padded to 16, stride 33
    int blk = blockIdx.x;                      // B*H*2 = 4096 blocks
    int b = blk >> 9;
    int rem = blk & 511;
    int h = rem >> 1;
    int wbase = (rem & 1) * 128;
    int t = threadIdx.x;

    // Stage 0: stage zero-padded conv2_w into LDS
    for (int i = t; i < 16 * 33; i += 256) {
        int n = i / 33, k = i - n * 33;
        bpad[i] = (n < 3 && k < 32) ? w2[n * C_ + k] : 0.f;
    }

    // Stage 1: depthwise 3x3 conv (zero pad) + bias, scaled by SE gate
    #pragma unroll
    for (int i = 0; i < 16; ++i) {
        int idx = i * 256 + t;                 // p fast dim -> coalesced loads
        int c = idx >> 7, p = idx & 127;
        int w = wbase + p;
        const float* plane = x + (size_t)(b * C_ + c) * N_;
        float acc = 0.f;
        #pragma unroll
        for (int dy = -1; dy <= 1; ++dy) {
            int hy = h + dy;
            if (hy < 0 || hy >= H_) continue;
            #pragma unroll
            for (int dx = -1; dx <= 1; ++dx) {
                int wx = w + dx;
                if (wx < 0 || wx >= W_) continue;
                acc += plane[hy * W_ + wx] * cw1[c*9 + (dy+1)*3 + (dx+1)];
            }
        }
        lds[p * 33 + c] = (acc + cb1[c]) * sgate[b * C_ + c];
    }
    __syncthreads();

    // Stage 2: LayerNorm over C + exact GELU, one pixel per thread
    if (t < 128) {
        float v[32];
        float mu = 0.f;
        #pragma unroll
        for (int c = 0; c < 32; ++c) { v[c] = lds[t*33 + c]; mu += v[c]; }
        mu *= (1.0f / 32.0f);
        float var = 0.f;
        #pragma unroll
        for (int c = 0; c < 32; ++c) { float d = v[c] - mu; var += d * d; }
        var *= (1.0f / 32.0f);
        float rs = rsqrtf(var + EPS_);
        #pragma unroll
        for (int c = 0; c < 32; ++c) {
            float xn = (v[c] - mu) * rs * lng[c] + lnb[c];
            lds[t*33 + c] = gelu_erf(xn);
        }
    }
    __syncthreads();

    // Stage 3: 1x1 conv [16 pix x 32ch] x [32ch x 3(padded 16)] via
    // 8 chained V_WMMA_F32_16X16X4_F32. EXEC all-1s here (no divergence).
    int lane  = t & 31;
    int wv    = t >> 5;                        // wave = pixel sub-tile
    int ln15  = lane & 15;                     // A: row M ; D: col N
    int khalf = lane >> 4;                     // 0 or 1
    v8f acc = {};
    #pragma unroll
    for (int kb = 0; kb < 32; kb += 4) {
        v2f a, bm;
        int k0 = kb + 2 * khalf;
        a.x  = lds[(wv*16 + ln15)*33 + k0];
        a.y  = lds[(wv*16 + ln15)*33 + k0 + 1];
        bm.x = bpad[ln15*33 + k0];
        bm.y = bpad[ln15*33 + k0 + 1];
        acc = __builtin_amdgcn_wmma_f32_16x16x4_f32(
                  false, a, false, bm, (short)0, acc, false, false);
    }

    // Extract D: lane column ln15 in {0,1,2} -> offsets / path position
    if (ln15 < 3) {
        #pragma unroll
        for (int r = 0; r < 8; ++r) {
            int pl  = wv*16 + khalf*8 + r;     // pixel within 128-tile
            int w   = wbase + pl;
            int pix = h * W_ + w;
            size_t o = (size_t)b * N_ + pix;
            float val = acc[r];
            if (ln15 == 0)
                posy[o] = tanhf(val) * (1.0f/255.0f) + ((h + 0.5f)*(1.0f/127.5f) - 1.0f);
            else if (ln15 == 1)
                posx[o] = tanhf(val) * (1.0f/255.0f) + ((w + 0.5f)*(1.0f/127.5f) - 1.0f);
            else
                path[o] = tanhf(val) + ((pix + 0.5f)*(2.0f/65535.0f) - 1.0f);
        }
    }
}

// ---------------- sort: iota + hybrid bitonic ----------------
__global__ void k_iota(int* __restrict__ vals) {
    int i = blockIdx.x * 256 + threadIdx.x;
    if (i < B_ * N_) vals[i] = i & (N_ - 1);
}

// LDS local sort: all stages k=2..CHUNK within a 2048-element chunk.
__global__ void k_bsort_local(float* __restrict__ keys, int* __restrict__ vals) {
    __shared__ float sk[CHUNK_];
    __shared__ int   sv[CHUNK_];
    size_t base = (size_t)blockIdx.x * CHUNK_;     // chunks never straddle batches
    int gbase = (int)(base & (N_ - 1));            // index within batch
    for (int i = threadIdx.x; i < CHUNK_; i += 256) {
        sk[i] = keys[base + i]; sv[i] = vals[base + i];
    }
    __syncthreads();
    for (int k = 2; k <= CHUNK_; k <<= 1) {
        for (int j = k >> 1; j > 0; j >>= 1) {
            for (int i = threadIdx.x; i < CHUNK_ / 2; i += 256) {
                int t   = 2 * i - (i & (j - 1));   // pair index with bit j clear
                int ixj = t | j;
                bool up = (((gbase + t) & k) == 0);
                float a = sk[t], bk = sk[ixj];
                bool sw = up ? (a > bk) : (a < bk);
                if (sw) {
                    sk[t] = bk; sk[ixj] = a;
                    int v = sv[t]; sv[t] = sv[ixj]; sv[ixj] = v;
                }
            }
            __syncthreads();
        }
    }
    for (int i = threadIdx.x; i < CHUNK_; i += 256) {
        keys[base + i] = sk[i]; vals[base + i] = sv[i];
    }
}

// Global pass for distance j >= CHUNK.
__global__ void k_bitonic(float* __restrict__ keys, int* __restrict__ vals,
                          int j, int k) {
    int i = blockIdx.x * 256 + threadIdx.x;    // B*N threads
    int b = i >> 16;
    int t = i & (N_ - 1);
    int ixj = t ^ j;
    if (ixj <= t) return;
    size_t base = (size_t)b << 16;
    float k0 = keys[base + t], k1 = keys[base + ixj];
    bool up = ((t & k) == 0);
    bool sw = up ? (k0 > k1) : (k0 < k1);
    if (sw) {
        keys[base + t] = k1; keys[base + ixj] = k0;
        int v0 = vals[base + t];
        vals[base + t] = vals[base + ixj];
        vals[base + ixj] = v0;
    }
}

// LDS merge: passes j = CHUNK/2 .. 1 of global stage k (k > CHUNK, so the
// direction bit is uniform per chunk).
__global__ void k_bmerge_local(float* __restrict__ keys, int* __restrict__ vals,
                               int k) {
    __shared__ float sk[CHUNK_];
    __shared__ int   sv[CHUNK_];
    size_t base = (size_t)blockIdx.x * CHUNK_;
    int gbase = (int)(base & (N_ - 1));
    bool up = ((gbase & k) == 0);                  // uniform for whole chunk
    for (int i = threadIdx.x; i < CHUNK_; i += 256) {
        sk[i] = keys[base + i]; sv[i] = vals[base + i];
    }
    __syncthreads();
    for (int j = CHUNK_ / 2; j > 0; j >>= 1) {
        for (int i = threadIdx.x; i < CHUNK_ / 2; i += 256) {
            int t   = 2 * i - (i & (j - 1));
            int ixj = t | j;
            float a = sk[t], bk = sk[ixj];
            bool sw = up ? (a > bk) : (a < bk);
            if (sw) {
                sk[t] = bk; sk[ixj] = a;
                int v = sv[t]; sv[t] = sv[ixj]; sv[ixj] = v;
            }
        }
        __syncthreads();
    }
    for (int i = threadIdx.x; i < CHUNK_; i += 256) {
        keys[base + i] = sk[i]; vals[base + i] = sv[i];
    }
}

// ---------------- fused sample + gather (zeros pad, align_corners=True) ----
__device__ __forceinline__ float bsample(const float* __restrict__ img,
                                         float gx, float gy) {
    float x0f = floorf(gx), y0f = floorf(gy);
    float wx = gx - x0f, wy = gy - y0f;
    int x0 = (int)x0f, y0 = (int)y0f;
    int x1 = x0 + 1,   y1 = y0 + 1;
    bool xi0 = (x0 >= 0) & (x0 < W_), yi0 = (y0 >= 0) & (y0 < H_);
    bool xi1 = (x1 >= 0) & (x1 < W_), yi1 = (y1 >= 0) & (y1 < H_);
    float r = 0.f;
    if (xi0 & yi0) r += img[y0*W_ + x0] * (1.f-wx)*(1.f-wy);
    if (xi1 & yi0) r += img[y0*W_ + x1] * wx*(1.f-wy);
    if (xi0 & yi1) r += img[y1*W_ + x0] * (1.f-wx)*wy;
    if (xi1 & yi1) r += img[y1*W_ + x1] * wx*wy;
    return r;
}

__global__ void k_gather(const float* __restrict__ xms, const float* __restrict__ xpan,
                         const float* __restrict__ rpe,
                         const float* __restrict__ posy, const float* __restrict__ posx,
                         const int* __restrict__ vals,
                         float* __restrict__ out) {
    int gid = blockIdx.x * 256 + threadIdx.x;  // B*N*C threads
    int c  = gid & 31;
    int nb = gid >> 5;                         // b*N + n
    int b  = nb >> 16;
    int j  = vals[nb];                         // sorted index
    size_t src = ((size_t)b << 16) + j;
    float py = posy[src], px = posx[src];
    int jh = j >> 8, jw = j & 255;
    float ry = (jh + 0.5f)*(1.0f/127.5f) - 1.0f;
    float rx = (jw + 0.5f)*(1.0f/127.5f) - 1.0f;
    float dx = (rx - px)*0.5f, dy = (ry - py)*0.5f;
    float gx  = (px + 1.0f)*127.5f, gy  = (py + 1.0f)*127.5f;
    float gbx = (dx + 1.0f)*127.5f, gby = (dy + 1.0f)*127.5f;
    float bias = bsample(rpe + (size_t)c * N_, gbx, gby);
    float vms  = bsample(xms  + (size_t)(b*C_ + c) * N_, gx, gy) + bias;
    float vpan = bsample(xpan + (size_t)(b*C_ + c) * N_, gx, gy) + bias;
    size_t o = (size_t)nb * C_ + c;            // [B,N,C] coalesced writes
    out[o] = vms;
    out[(size_t)B_*N_*C_ + o] = vpan;
    if (c == 0) {
        int* idxout = (int*)(out + 2*(size_t)B_*N_*C_);
        idxout[nb] = j;
    }
}

// ---------------- launcher ----------------
extern "C" void kernel_launch(void* const* d_in, const int* in_sizes, int n_in,
                              void* d_out, int out_size, void* d_ws, size_t ws_size,
                              hipStream_t stream) {
    (void)in_sizes; (void)n_in; (void)out_size; (void)ws_size;
    const float* x_ms   = (const float*)d_in[0];
    const float* x_pan  = (const float*)d_in[1];
    const float* conv1w = (const float*)d_in[2];
    const float* conv1b = (const float*)d_in[3];
    const float* caw1   = (const float*)d_in[4];
    const float* cab1   = (const float*)d_in[5];
    const float* caw2   = (const float*)d_in[6];
    const float* cab2   = (const float*)d_in[7];
    const float* lng    = (const float*)d_in[8];
    const float* lnb    = (const float*)d_in[9];
    const float* conv2w = (const float*)d_in[10];
    const float* rpetab = (const float*)d_in[11];
    float* out = (float*)d_out;

    // workspace layout (floats)
    float* ws    = (float*)d_ws;
    float* posy  = ws;                         // B*N
    float* posx  = ws + 1 * (B_*N_);           // B*N
    float* path  = ws + 2 * (B_*N_);           // B*N (sort keys, in place)
    int*   vals  = (int*)(ws + 3 * (B_*N_));   // B*N
    float* rpe   = ws + 4 * (B_*N_);           // C*N
    float* xc    = rpe + (size_t)C_ * N_;      // B*C
    float* sgate = xc + B_*C_;                 // B*C

    k_mean<<<B_*C_, 256, 0, stream>>>(x_ms, xc);
    k_gate<<<1, 256, 0, stream>>>(xc, caw1, cab1, caw2, cab2, sgate);
    k_rpe<<<(C_*N_)/256, 256, 0, stream>>>(rpetab, rpe);
    k_offset<<<B_*H_*(W_/128), 256, 0, stream>>>(x_ms, conv1w, conv1b, sgate,
                                                 lng, lnb, conv2w,
                                                 posy, posx, path);
    k_iota<<<(B_*N_)/256, 256, 0, stream>>>(vals);

    // Hybrid bitonic argsort: 1 local-sort + 15 global passes + 5 local merges
    int nchunks = (B_ * N_) / CHUNK_;          // 256 blocks
    k_bsort_local<<<nchunks, 256, 0, stream>>>(path, vals);
    for (int k = CHUNK_ * 2; k <= N_; k <<= 1) {
        for (int j = k >> 1; j >= CHUNK_; j >>= 1)
            k_bitonic<<<(B_*N_)/256, 256, 0, stream>>>(path, vals, j, k);
        k_bmerge_local<<<nchunks, 256, 0, stream>>>(path, vals, k);
    }

    k_gather<<<(B_*N_*C_)/256, 256, 0, stream>>>(x_ms, x_pan, rpe, posy, posx,
                                                 vals, out);
}